// LinkPredModelMultiOutput_47699906789908
// MI455X (gfx1250) — compile-verified
//
#include <hip/hip_runtime.h>

// ---------------------------------------------------------------------------
// GraphSAGE (2 layers) + link scorer for MI455X (gfx1250, wave32).
// All GEMMs use V_WMMA_F32_16X16X4_F32 (exact fp32 tensor-core path).
// Scatter aggregation uses GLOBAL_ATOMIC_ADD_F32 (no-return -> STOREcnt path).
// A-operands staged through padded LDS (conflict-free, ds_load_2addr pairs).
// ---------------------------------------------------------------------------

typedef __attribute__((ext_vector_type(2))) float v2f;
typedef __attribute__((ext_vector_type(8))) float v8f;

#define WMMA_F32(a, b, c) \
  __builtin_amdgcn_wmma_f32_16x16x4_f32(false, (a), false, (b), (short)0, (c), false, false)

// ---------------------------------------------------------------------------
__global__ __launch_bounds__(256) void fill_zero_f4(float4* __restrict__ p, int n4) {
  int i = blockIdx.x * blockDim.x + threadIdx.x;
  if (i < n4) p[i] = make_float4(0.f, 0.f, 0.f, 0.f);
}

__global__ __launch_bounds__(256) void degree_kernel(const int* __restrict__ dst,
                                                     float* __restrict__ deg, int E) {
  int e = blockIdx.x * blockDim.x + threadIdx.x;
  if (e < E) atomicAdd(&deg[dst[e]], 1.0f);
}

__global__ __launch_bounds__(256) void invert_deg(float* __restrict__ deg, int N) {
  int i = blockIdx.x * blockDim.x + threadIdx.x;
  if (i < N) deg[i] = 1.0f / fmaxf(deg[i], 1.0f);
}

// Scatter-add node rows along edges: agg[dst[e]][:] += X[src[e]][:]
// F consecutive threads handle one edge -> coalesced row read + row atomics.
template <int F>
__global__ __launch_bounds__(256) void scatter_add(const float* __restrict__ X,
                                                   const int* __restrict__ src,
                                                   const int* __restrict__ dst,
                                                   float* __restrict__ agg, int E) {
  long long idx = (long long)blockIdx.x * blockDim.x + threadIdx.x;
  if (idx >= (long long)E * F) return;
  constexpr int SH = (F == 128) ? 7 : 8;
  int e = (int)(idx >> SH);
  int f = (int)(idx & (F - 1));
  int s = src[e];
  int d = dst[e];
  atomicAdd(&agg[(size_t)d * F + f], X[(size_t)s * F + f]);
}

// ---------------------------------------------------------------------------
// Y[m, n] = act( X[m,:]@Wself[:,n] + (AGG[m,:]*invdeg[m])@Wneigh[:,n] + bias[n] )
// Block: 256 threads = 8 waves. Block covers 16 rows x 128 cols (wave: 16x16).
// ---------------------------------------------------------------------------
template <int K, int NOUT, bool RELU>
__global__ __launch_bounds__(256) void sage_gemm(const float* __restrict__ X,
                                                 const float* __restrict__ AGG,
                                                 const float* __restrict__ invdeg,
                                                 const float* __restrict__ Wself,
                                                 const float* __restrict__ Wneigh,
                                                 const float* __restrict__ bias,
                                                 float* __restrict__ Y, int N) {
  constexpr int SP = K + 4;   // pad: row*SP % 64 == row*4 -> conflict-free column reads
  constexpr int KV = K / 4;
  __shared__ float sX[16 * SP];
  __shared__ float sA[16 * SP];

  const int m0  = blockIdx.x * 16;
  const int tid = threadIdx.x;

  // Cooperative staging of the 16xK self tile and the inv-deg-scaled agg tile.
  for (int i = tid; i < 16 * KV; i += 256) {
    int row = i / KV;
    int j   = i - row * KV;
    float4 xv = reinterpret_cast<const float4*>(X)[(size_t)(m0 + row) * KV + j];
    *reinterpret_cast<float4*>(&sX[row * SP + 4 * j]) = xv;
    float inv = invdeg[m0 + row];
    float4 av = reinterpret_cast<const float4*>(AGG)[(size_t)(m0 + row) * KV + j];
    av.x *= inv; av.y *= inv; av.z *= inv; av.w *= inv;
    *reinterpret_cast<float4*>(&sA[row * SP + 4 * j]) = av;
    // Warm L2 for the next row-block's tiles (node matrices are L2-resident).
    if (m0 + 16 + row < N) {
      __builtin_prefetch(&X[(size_t)(m0 + 16 + row) * K + 4 * j], 0, 3);
      __builtin_prefetch(&AGG[(size_t)(m0 + 16 + row) * K + 4 * j], 0, 3);
    }
  }
  __syncthreads();

  const int lane = tid & 31;
  const int wave = tid >> 5;
  const int n0   = blockIdx.y * 128 + wave * 16;
  const int col  = lane & 15;              // N index (B / C operands)
  const int koff = (lane >> 4) << 1;       // K split: lanes 0-15 -> K{0,1}, 16-31 -> K{2,3}
  const float* sxr = &sX[(lane & 15) * SP];
  const float* sar = &sA[(lane & 15) * SP];

  v8f c = {0.f, 0.f, 0.f, 0.f, 0.f, 0.f, 0.f, 0.f};

#pragma unroll 4
  for (int k = 0; k < K; k += 4) {
    const int kk = k + koff;
    v2f a1; a1.x = sxr[kk];  a1.y = sxr[kk + 1];
    v2f b1; b1.x = Wself[(size_t)kk * NOUT + n0 + col];
            b1.y = Wself[(size_t)(kk + 1) * NOUT + n0 + col];
    c = WMMA_F32(a1, b1, c);
    v2f a2; a2.x = sar[kk];  a2.y = sar[kk + 1];
    v2f b2; b2.x = Wneigh[(size_t)kk * NOUT + n0 + col];
            b2.y = Wneigh[(size_t)(kk + 1) * NOUT + n0 + col];
    c = WMMA_F32(a2, b2, c);
  }

  const float bcol = bias[n0 + col];
#pragma unroll
  for (int i = 0; i < 8; ++i) {
    int row = m0 + i + ((lane >> 4) << 3);   // C layout: vgpr i -> M = i + 8*(lane/16)
    float v = c[i] + bcol;
    if (RELU) v = fmaxf(v, 0.0f);
    Y[(size_t)row * NOUT + n0 + col] = v;    // lanes 0-15 write contiguous 64B
  }
}

// ---------------------------------------------------------------------------
// score[e, c] = h2[src[e]]@Wu[:,c] + h2[dst[e]]@Wv[:,c] + bp[c]
// One wave = 16 edges x 16 classes tile; Wu/Wv/bp staged in LDS.
// Full tiles take a wave-uniform unguarded path (coalesced 64B stores).
// ---------------------------------------------------------------------------
__global__ __launch_bounds__(256) void score_kernel(const float* __restrict__ H,
                                                    const int* __restrict__ src,
                                                    const int* __restrict__ dst,
                                                    const float* __restrict__ Wp,
                                                    const float* __restrict__ bp,
                                                    float* __restrict__ out, int E) {
  __shared__ float sW[4096];   // [0,2048): Wu (128x16), [2048,4096): Wv (128x16)
  __shared__ float sbp[16];
  const int tid = threadIdx.x;
  for (int i = tid; i < 4096; i += 256) sW[i] = Wp[i];
  if (tid < 16) sbp[tid] = bp[tid];
  __syncthreads();

  const int lane = tid & 31;
  const int wave = tid >> 5;
  const int e0   = (blockIdx.x * 8 + wave) * 16;
  if (e0 >= E) return;                       // wave-uniform exit (after barrier)
  const bool full = (e0 + 16 <= E);          // wave-uniform

  int e = e0 + (lane & 15);
  if (!full && e >= E) e = E - 1;            // tail-only clamp (EXEC restored before WMMA)
  const int s  = src[e];
  const int d2 = dst[e];
  const float* hs = H + (size_t)s * 128;
  const float* hd = H + (size_t)d2 * 128;
  const int koff = (lane >> 4) << 1;
  const int col  = lane & 15;

  v8f c = {0.f, 0.f, 0.f, 0.f, 0.f, 0.f, 0.f, 0.f};

#pragma unroll 4
  for (int k = 0; k < 128; k += 4) {
    const int kk = k + koff;
    v2f a1; a1.x = hs[kk]; a1.y = hs[kk + 1];
    v2f b1; b1.x = sW[kk * 16 + col]; b1.y = sW[(kk + 1) * 16 + col];
    c = WMMA_F32(a1, b1, c);
    v2f a2; a2.x = hd[kk]; a2.y = hd[kk + 1];
    v2f b2; b2.x = sW[2048 + kk * 16 + col]; b2.y = sW[2048 + (kk + 1) * 16 + col];
    c = WMMA_F32(a2, b2, c);
  }

  const float bc = sbp[col];
  if (full) {
#pragma unroll
    for (int i = 0; i < 8; ++i) {
      int ee = e0 + i + ((lane >> 4) << 3);
      out[(size_t)ee * 16 + col] = c[i] + bc;   // unguarded, coalesced
    }
  } else {
#pragma unroll
    for (int i = 0; i < 8; ++i) {
      int ee = e0 + i + ((lane >> 4) << 3);
      if (ee < E) out[(size_t)ee * 16 + col] = c[i] + bc;
    }
  }
}

// ---------------------------------------------------------------------------
extern "C" void kernel_launch(void* const* d_in, const int* in_sizes, int n_in,
                              void* d_out, int out_size, void* d_ws, size_t ws_size,
                              hipStream_t stream) {
  const float* x   = (const float*)d_in[0];
  const int*   src = (const int*)d_in[1];
  const int*   dst = (const int*)d_in[2];
  const float* W1s = (const float*)d_in[3];
  const float* W1n = (const float*)d_in[4];
  const float* b1  = (const float*)d_in[5];
  const float* W2s = (const float*)d_in[6];
  const float* W2n = (const float*)d_in[7];
  const float* b2  = (const float*)d_in[8];
  const float* Wp  = (const float*)d_in[9];
  const float* bp  = (const float*)d_in[10];
  float* out = (float*)d_out;

  const int E = in_sizes[1];
  const int N = in_sizes[0] / 128;   // 100000

  // Workspace layout (floats): invdeg[N] | agg[N*256] | h1[N*256] | h2[N*128]
  float* ws     = (float*)d_ws;
  float* invdeg = ws;
  float* agg    = ws + N;
  float* h1     = agg + (size_t)N * 256;
  float* h2     = h1 + (size_t)N * 256;

  // ---- degrees -> inverse degrees ----
  fill_zero_f4<<<(N / 4 + 255) / 256, 256, 0, stream>>>((float4*)invdeg, N / 4);
  degree_kernel<<<(E + 255) / 256, 256, 0, stream>>>(dst, invdeg, E);
  invert_deg<<<(N + 255) / 256, 256, 0, stream>>>(invdeg, N);

  // ---- layer 1: aggregate x, then fused GEMM + bias + relu ----
  {
    int n4 = N * 128 / 4;
    fill_zero_f4<<<(n4 + 255) / 256, 256, 0, stream>>>((float4*)agg, n4);
    long long total = (long long)E * 128;
    int blocks = (int)((total + 255) / 256);
    scatter_add<128><<<blocks, 256, 0, stream>>>(x, src, dst, agg, E);
    sage_gemm<128, 256, true><<<dim3(N / 16, 2), 256, 0, stream>>>(
        x, agg, invdeg, W1s, W1n, b1, h1, N);
  }

  // ---- layer 2: aggregate h1, fused GEMM + bias (no relu) ----
  {
    int n4 = N * 256 / 4;
    fill_zero_f4<<<(n4 + 255) / 256, 256, 0, stream>>>((float4*)agg, n4);
    long long total = (long long)E * 256;
    int blocks = (int)((total + 255) / 256);
    scatter_add<256><<<blocks, 256, 0, stream>>>(h1, src, dst, agg, E);
    sage_gemm<256, 128, false><<<dim3(N / 16, 1), 256, 0, stream>>>(
        h1, agg, invdeg, W2s, W2n, b2, h2, N);
  }

  // ---- edge scores ----
  {
    int tiles = (E + 15) / 16;
    int blocks = (tiles + 7) / 8;
    score_kernel<<<blocks, 256, 0, stream>>>(h2, src, dst, Wp, bp, out, E);
  }
}